// ProtoNetNLP_23270132809995
// MI455X (gfx1250) — compile-verified
//
#include <hip/hip_runtime.h>

// Episode constants (match reference)
#define D_DIM   768
#define N_WAYS  10
#define NPAD    16      // prototypes padded to 16 rows (zeros) for the WMMA N-tile
#define K_SHOTS 5
#define Q_TOT   500
#define B_EP    8
#define NTILES  32      // ceil(500/16) query tiles per episode

typedef __attribute__((ext_vector_type(2))) float v2f;
typedef __attribute__((ext_vector_type(8))) float v8f;

// ---------------------------------------------------------------------------
// Kernel 1: prototypes = mean over K shots, written into zero-padded [B,16,D]
// ---------------------------------------------------------------------------
__global__ void proto_mean_kernel(const float* __restrict__ support,
                                  float* __restrict__ protos) {
  const int bn = blockIdx.x;          // 0 .. B*NPAD-1
  const int b  = bn >> 4;
  const int n  = bn & 15;
  float* out = protos + (size_t)bn * D_DIM;
  if (n >= N_WAYS) {                  // zero padding rows -> zero dot, zero norm
    for (int d = threadIdx.x; d < D_DIM; d += blockDim.x) out[d] = 0.0f;
    return;
  }
  const float* base = support + ((size_t)(b * N_WAYS + n) * K_SHOTS) * D_DIM;
  for (int d = threadIdx.x; d < D_DIM; d += blockDim.x) {
    float s = 0.0f;
#pragma unroll
    for (int k = 0; k < K_SHOTS; ++k) s += base[(size_t)k * D_DIM + d];
    out[d] = s * (1.0f / (float)K_SHOTS);
  }
}

// ---------------------------------------------------------------------------
// Kernel 2: logits[b,q,n] = 2*(q.p) - ||q||^2 - ||p||^2 via v_wmma_f32_16x16x4_f32
// One wave computes a 16(query) x 16(proto) tile, K accumulated in steps of 4.
// ---------------------------------------------------------------------------
__global__ void __launch_bounds__(256, 1)
dist_wmma_kernel(const float* __restrict__ query,
                 const float* __restrict__ protos,
                 float* __restrict__ logits) {
  const int wave = threadIdx.x >> 5;
  const int lane = threadIdx.x & 31;
  const int tile = blockIdx.x * 8 + wave;     // 0 .. B*NTILES-1
  const int b    = tile >> 5;                 // NTILES = 32 tiles per episode
  const int t    = tile & 31;
  const int q0   = t * 16;

  // A-matrix (16x4 f32) lane mapping: row = lane&15, K-pair offset = 2*(lane>>4).
  // B-matrix (4x16 f32) is the transposed identical pattern on the proto rows.
  const int row  = lane & 15;
  const int koff = (lane >> 4) << 1;

  int qrow = q0 + row;
  if (qrow >= Q_TOT) qrow = Q_TOT - 1;        // clamp; results for pads unstored
  const float* qptr = query  + ((size_t)b * Q_TOT + qrow) * D_DIM + koff;
  const float* pptr = protos + ((size_t)b * NPAD + row)  * D_DIM + koff;

  v8f c = {};
  float qn2p = 0.0f, pn2p = 0.0f;             // fused norm partials
#pragma unroll 8
  for (int k = 0; k < D_DIM; k += 4) {
    v2f a  = *(const v2f*)(qptr + k);
    v2f bb = *(const v2f*)(pptr + k);
    qn2p += a.x * a.x + a.y * a.y;
    pn2p += bb.x * bb.x + bb.y * bb.y;
    // 8 args: (neg_a, A, neg_b, B, c_mod, C, reuse_a, reuse_b)
    c = __builtin_amdgcn_wmma_f32_16x16x4_f32(false, a, false, bb,
                                              (short)0, c, false, false);
  }

  // Combine half-wave partials: lanes l and l+16 cover the same row/col.
  float pn2  = pn2p + __shfl_xor(pn2p, 16, 32);   // lane holds ||p[lane&15]||^2
  float qn2s = qn2p + __shfl_xor(qn2p, 16, 32);   // lane holds ||q[lane&15]||^2

  // C/D layout: lane -> n = lane&15; VGPR r -> m = r + 8*(lane>>4)
  const int n     = lane & 15;
  const int mbase = (lane >> 4) * 8;
#pragma unroll
  for (int r = 0; r < 8; ++r) {
    const int m   = mbase + r;
    float qn2     = __shfl(qn2s, m, 32);          // ||q[m]||^2 lives in lane m
    float val     = 2.0f * c[r] - qn2 - pn2;      // = -||q - p||^2
    if (q0 + m < Q_TOT)
      logits[(((size_t)b * Q_TOT) + (q0 + m)) * NPAD + n] = val;
  }
}

// ---------------------------------------------------------------------------
// Kernel 3: min / concat(min-1) / softmax / argmax / max-prob, one thread/row
// d_out = [ fh: rows*11 | ph: rows*11 | yh_top: rows | ph_top: rows ]
// ---------------------------------------------------------------------------
__global__ void finalize_kernel(const float* __restrict__ logits,
                                float* __restrict__ out) {
  const int ROWS = B_EP * Q_TOT;
  const int i = blockIdx.x * blockDim.x + threadIdx.x;
  if (i >= ROWS) return;

  const float* lrow = logits + (size_t)i * NPAD;
  float l[N_WAYS + 1];
  float mn = lrow[0];
#pragma unroll
  for (int n = 0; n < N_WAYS; ++n) { l[n] = lrow[n]; mn = fminf(mn, l[n]); }
  l[N_WAYS] = mn - 1.0f;

  float mx = l[0]; int arg = 0;
#pragma unroll
  for (int n = 1; n <= N_WAYS; ++n)
    if (l[n] > mx) { mx = l[n]; arg = n; }     // first-max tie-break, like argmax

  float p[N_WAYS + 1];
  float sum = 0.0f;
#pragma unroll
  for (int n = 0; n <= N_WAYS; ++n) { p[n] = __expf(l[n] - mx); sum += p[n]; }
  const float inv = 1.0f / sum;

  float* fh   = out;
  float* ph   = out + (size_t)ROWS * (N_WAYS + 1);
  float* yh   = out + 2 * (size_t)ROWS * (N_WAYS + 1);
  float* ptop = yh + ROWS;

  float pmax = 0.0f;
#pragma unroll
  for (int n = 0; n <= N_WAYS; ++n) {
    fh[(size_t)i * (N_WAYS + 1) + n] = l[n];
    const float pv = p[n] * inv;
    ph[(size_t)i * (N_WAYS + 1) + n] = pv;
    pmax = fmaxf(pmax, pv);
  }
  yh[i]   = (float)arg;
  ptop[i] = pmax;
}

// ---------------------------------------------------------------------------
extern "C" void kernel_launch(void* const* d_in, const int* in_sizes, int n_in,
                              void* d_out, int out_size, void* d_ws, size_t ws_size,
                              hipStream_t stream) {
  const float* support = (const float*)d_in[0];  // [B*N*K, D] f32
  const float* query   = (const float*)d_in[1];  // [B*Q,   D] f32

  // Workspace layout: padded prototypes, then padded logits.
  float* protos = (float*)d_ws;                                  // B*16*768 f32
  float* logits = (float*)((char*)d_ws +
                  (size_t)B_EP * NPAD * D_DIM * sizeof(float));  // B*500*16 f32

  proto_mean_kernel<<<B_EP * NPAD, 256, 0, stream>>>(support, protos);
  dist_wmma_kernel<<<(B_EP * NTILES) / 8, 256, 0, stream>>>(query, protos, logits);
  finalize_kernel<<<(B_EP * Q_TOT + 255) / 256, 256, 0, stream>>>(logits,
                                                                  (float*)d_out);
}